// MZILinear_11785390260443
// MI455X (gfx1250) — compile-verified
//
#include <hip/hip_runtime.h>

#define NN 512
#define NPAIRS ((NN * (NN - 1)) / 2)   // 130816

typedef float v2f __attribute__((ext_vector_type(2)));
typedef float v8f __attribute__((ext_vector_type(8)));

// ---------------------------------------------------------------------------
// Kernel 0: precompute cos/sin of all Givens angles for U and V.
// ---------------------------------------------------------------------------
__global__ void mzi_sincos(const float* __restrict__ phiU,
                           const float* __restrict__ phiV,
                           float* __restrict__ cU, float* __restrict__ sU,
                           float* __restrict__ cV, float* __restrict__ sV) {
    int i = blockIdx.x * blockDim.x + threadIdx.x;
    if (i < NPAIRS) {
        float s, c;
        sincosf(phiU[i], &s, &c);
        cU[i] = c; sU[i] = s;
    } else if (i < 2 * NPAIRS) {
        int k = i - NPAIRS;
        float s, c;
        sincosf(phiV[k], &s, &c);
        cV[k] = c; sV[k] = s;
    }
}

// ---------------------------------------------------------------------------
// Kernel 1: reconstruct U (blockIdx.y==0) and V (blockIdx.y==1).
// Columns are independent under row-rotations; each thread owns one column
// held in LDS (stride 513 -> conflict-free). Pivot row value chained in a
// register across each Reck inner loop; (c,s) staged to LDS in chunks and
// consumed as broadcasts.
// ---------------------------------------------------------------------------
#define RCOLS 64
#define RSTR  513
#define RCH   1024   // power of two

__global__ void mzi_reconstruct(const float* __restrict__ cU, const float* __restrict__ sU,
                                const float* __restrict__ dU, float* __restrict__ Uo,
                                const float* __restrict__ cV, const float* __restrict__ sV,
                                const float* __restrict__ dV, float* __restrict__ Vo) {
    extern __shared__ float smem[];
    float* col = smem;                    // RCOLS * RSTR
    float* chC = smem + RCOLS * RSTR;     // RCH
    float* chS = chC + RCH;               // RCH

    const float* cArr; const float* sArr; const float* del; float* outM;
    if (blockIdx.y == 0) { cArr = cU; sArr = sU; del = dU; outM = Uo; }
    else                 { cArr = cV; sArr = sV; del = dV; outM = Vo; }

    const int t    = threadIdx.x;
    const int colG = blockIdx.x * RCOLS + t;
    float* myc = col + t * RSTR;

    // U0 = I : column colG is e_colG
    for (int r = 0; r < NN; ++r) myc[r] = (r == colG) ? 1.0f : 0.0f;

    int k = 0;  // global rotation index, uniform across the block
    for (int i = 0; i < NN - 1; ++i) {
        float rp = myc[i];                       // pivot row value, chained
        for (int j = NN - 1; j > i; --j) {
            if ((k & (RCH - 1)) == 0) {          // refill (c,s) chunk
                __syncthreads();
                for (int u = t; u < RCH; u += blockDim.x) {
                    int g = k + u;
                    if (g < NPAIRS) { chC[u] = cArr[g]; chS[u] = sArr[g]; }
                }
                __syncthreads();
            }
            const int kc = k & (RCH - 1);
            const float cc = chC[kc];
            const float ss = chS[kc];
            ++k;
            const float rq  = myc[j];
            const float nrp = fmaf(cc, rp, -ss * rq);
            myc[j]          = fmaf(ss, rp,  cc * rq);
            rp = nrp;
        }
        myc[i] = rp;                             // row i retired
    }
    __syncthreads();

    // Write out with diag(delta) row scaling; coalesced across columns.
    for (int r = 0; r < NN; ++r)
        outM[r * NN + colG] = del[r] * col[t * RSTR + r];
}

// ---------------------------------------------------------------------------
// Kernel 2: Wt[k][n] = sum_j sigma[j] * V[j][k] * U[n][j]   (= W^T, 512x512)
// ---------------------------------------------------------------------------
__global__ void mzi_weight(const float* __restrict__ U, const float* __restrict__ Vm,
                           const float* __restrict__ sigma, float* __restrict__ Wt) {
    __shared__ float Vs[16][17];
    __shared__ float Us[16][17];
    __shared__ float sg[16];
    const int tx = threadIdx.x, ty = threadIdx.y;
    const int kBase = blockIdx.y * 16, nBase = blockIdx.x * 16;
    float acc = 0.0f;
    for (int j0 = 0; j0 < NN; j0 += 16) {
        Vs[ty][tx] = Vm[(j0 + ty) * NN + kBase + tx];   // V[j][k]
        Us[ty][tx] = U[(nBase + ty) * NN + j0 + tx];    // U[n][j]
        if (ty == 0) sg[tx] = sigma[j0 + tx];
        __syncthreads();
#pragma unroll
        for (int jj = 0; jj < 16; ++jj)
            acc = fmaf(sg[jj] * Vs[jj][ty], Us[tx][jj], acc);
        __syncthreads();
    }
    Wt[(kBase + ty) * NN + nBase + tx] = acc;
}

// ---------------------------------------------------------------------------
// Kernel 3: out[M=32768][512] = X[M][512] @ Wt[512][512]  via fp32 WMMA.
// Block tile 128x128, 8 waves, each wave 32x64 (2x4 fragments of 16x16).
// A staged transposed [k][m], B staged [k][n]; stride 136 for conflict-free
// half-wave fragment reads (rows k and k+2 land on disjoint banks).
// ---------------------------------------------------------------------------
#define BM 128
#define BN 128
#define KC 32
#define BMP 136
#define BNP 136

__global__ __launch_bounds__(256) void mzi_gemm(const float* __restrict__ X,
                                                const float* __restrict__ Wt,
                                                float* __restrict__ out) {
    __shared__ float Ast[KC][BMP];
    __shared__ float Bs[KC][BNP];

    const int tid  = threadIdx.x;
    const int lane = tid & 31;
    const int wave = tid >> 5;
    const int wm   = wave & 3;        // 4 waves along M
    const int wn   = wave >> 2;       // 2 waves along N
    const int half = lane >> 4;       // 0: lanes 0-15, 1: lanes 16-31
    const int l16  = lane & 15;
    const int mBase = blockIdx.y * BM;
    const int nBase = blockIdx.x * BN;

    const v8f vzero = {0.f, 0.f, 0.f, 0.f, 0.f, 0.f, 0.f, 0.f};
    v8f acc[2][4];
#pragma unroll
    for (int i = 0; i < 2; ++i)
#pragma unroll
        for (int j = 0; j < 4; ++j) acc[i][j] = vzero;

    for (int k0 = 0; k0 < NN; k0 += KC) {
        // Stage A tile (BM x KC) transposed into Ast[k][m]; coalesced float4 reads.
        for (int s = tid; s < BM * KC / 4; s += 256) {
            const int row = s >> 3;        // 0..127
            const int kq  = s & 7;         // 0..7 (groups of 4 k)
            const float4 v = *reinterpret_cast<const float4*>(
                &X[(mBase + row) * NN + k0 + kq * 4]);
            Ast[kq * 4 + 0][row] = v.x;
            Ast[kq * 4 + 1][row] = v.y;
            Ast[kq * 4 + 2][row] = v.z;
            Ast[kq * 4 + 3][row] = v.w;
        }
        // Stage B tile (KC x BN) into Bs[k][n]; coalesced float4 read+write.
        for (int s = tid; s < KC * BN / 4; s += 256) {
            const int kr = s >> 5;         // 0..31
            const int nq = s & 31;         // 0..31 (groups of 4 n)
            const float4 v = *reinterpret_cast<const float4*>(
                &Wt[(k0 + kr) * NN + nBase + nq * 4]);
            *reinterpret_cast<float4*>(&Bs[kr][nq * 4]) = v;
        }
        __syncthreads();

#pragma unroll
        for (int kk = 0; kk < KC; kk += 4) {
            // A fragments: 16x4 f32 -> v2f; x: K=kk+2*half, y: K=kk+2*half+1
            const int m0 = wm * 32 + l16;
            v2f a0, a1;
            a0.x = Ast[kk + 2 * half][m0];
            a0.y = Ast[kk + 2 * half + 1][m0];
            a1.x = Ast[kk + 2 * half][m0 + 16];
            a1.y = Ast[kk + 2 * half + 1][m0 + 16];
            // B fragments: 4x16 f32 -> v2f, same K striping
            const int n0 = wn * 64 + l16;
            v2f b0, b1, b2, b3;
            b0.x = Bs[kk + 2 * half][n0];      b0.y = Bs[kk + 2 * half + 1][n0];
            b1.x = Bs[kk + 2 * half][n0 + 16]; b1.y = Bs[kk + 2 * half + 1][n0 + 16];
            b2.x = Bs[kk + 2 * half][n0 + 32]; b2.y = Bs[kk + 2 * half + 1][n0 + 32];
            b3.x = Bs[kk + 2 * half][n0 + 48]; b3.y = Bs[kk + 2 * half + 1][n0 + 48];

            acc[0][0] = __builtin_amdgcn_wmma_f32_16x16x4_f32(false, a0, false, b0, (short)0, acc[0][0], false, false);
            acc[0][1] = __builtin_amdgcn_wmma_f32_16x16x4_f32(false, a0, false, b1, (short)0, acc[0][1], false, false);
            acc[0][2] = __builtin_amdgcn_wmma_f32_16x16x4_f32(false, a0, false, b2, (short)0, acc[0][2], false, false);
            acc[0][3] = __builtin_amdgcn_wmma_f32_16x16x4_f32(false, a0, false, b3, (short)0, acc[0][3], false, false);
            acc[1][0] = __builtin_amdgcn_wmma_f32_16x16x4_f32(false, a1, false, b0, (short)0, acc[1][0], false, false);
            acc[1][1] = __builtin_amdgcn_wmma_f32_16x16x4_f32(false, a1, false, b1, (short)0, acc[1][1], false, false);
            acc[1][2] = __builtin_amdgcn_wmma_f32_16x16x4_f32(false, a1, false, b2, (short)0, acc[1][2], false, false);
            acc[1][3] = __builtin_amdgcn_wmma_f32_16x16x4_f32(false, a1, false, b3, (short)0, acc[1][3], false, false);
        }
        __syncthreads();
    }

    // Store: C/D layout — VGPR v holds M = v + 8*half, N = l16.
#pragma unroll
    for (int i = 0; i < 2; ++i) {
#pragma unroll
        for (int j = 0; j < 4; ++j) {
            const int row0 = mBase + wm * 32 + i * 16 + 8 * half;
            const int colI = nBase + wn * 64 + j * 16 + l16;
#pragma unroll
            for (int v = 0; v < 8; ++v)
                out[(row0 + v) * NN + colI] = acc[i][j][v];
        }
    }
}

// ---------------------------------------------------------------------------
extern "C" void kernel_launch(void* const* d_in, const int* in_sizes, int n_in,
                              void* d_out, int out_size, void* d_ws, size_t ws_size,
                              hipStream_t stream) {
    const float* x     = (const float*)d_in[0];   // [16,2048,512]
    const float* phiU  = (const float*)d_in[1];   // [130816]
    const float* delU  = (const float*)d_in[2];   // [512]
    const float* phiV  = (const float*)d_in[3];   // [130816]
    const float* delV  = (const float*)d_in[4];   // [512]
    const float* sigma = (const float*)d_in[5];   // [512]
    float* out = (float*)d_out;                   // [32768,512]

    float* ws = (float*)d_ws;
    float* cU = ws;
    float* sU = cU + NPAIRS;
    float* cV = sU + NPAIRS;
    float* sV = cV + NPAIRS;
    float* U  = sV + NPAIRS;          // 512*512
    float* V  = U + NN * NN;          // 512*512
    float* Wt = V + NN * NN;          // 512*512

    // 0) angles -> (cos, sin)
    const int tot = 2 * NPAIRS;
    mzi_sincos<<<(tot + 255) / 256, 256, 0, stream>>>(phiU, phiV, cU, sU, cV, sV);

    // 1) Givens reconstruction of U and V (column-parallel LDS scan)
    const size_t ldsB = (size_t)(RCOLS * RSTR + 2 * RCH) * sizeof(float);
    mzi_reconstruct<<<dim3(NN / RCOLS, 2), RCOLS, ldsB, stream>>>(
        cU, sU, delU, U, cV, sV, delV, V);

    // 2) Wt = V^T diag(sigma) U^T  (512x512)
    mzi_weight<<<dim3(NN / 16, NN / 16), dim3(16, 16), 0, stream>>>(U, V, sigma, Wt);

    // 3) out = x @ Wt  (fp32 WMMA GEMM)
    mzi_gemm<<<dim3(NN / BN, (16 * 2048) / BM), 256, 0, stream>>>(x, Wt, out);
}